// PopulationSystemNormalizedSplit_45913200394314
// MI455X (gfx1250) — compile-verified
//
#include <hip/hip_runtime.h>
#include <hip/hip_bf16.h>
#include <math.h>

typedef float v2f __attribute__((ext_vector_type(2)));
typedef float v8f __attribute__((ext_vector_type(8)));

namespace {
constexpr int    MM        = 8192;
constexpr int    NMAX      = 16;          // hard upper bound on substeps (CFL <= 14)
constexpr double D_DM      = 0.9 / 8192.0;
constexpr float  F_DM      = (float)D_DM;
constexpr float  F_INV_DM  = (float)(8192.0 / 0.9);
constexpr float  F_KS_MAX  = 0.8f;
constexpr float  F_K_S     = 2.0f;
constexpr float  F_S_MAX   = 10.0f;
constexpr float  F_KAPPA   = 0.1f;        // N_MAX / S_MAX
constexpr float  F_M_MAX   = 1.0f;
constexpr float  F_M_MIN   = 0.45f;
constexpr float  F_EPS     = 1e-12f;
}

// ---------------------------------------------------------------------------
// Kernel 0: compute N (substep count) and h = DT/N, in double like the
// reference's numpy path. m is monotone increasing so both maxima sit at m[M-1].
// ---------------------------------------------------------------------------
__global__ void compute_substeps_k(const float* __restrict__ x,
                                   const float* __restrict__ m,
                                   int* __restrict__ ws_N,
                                   float* __restrict__ ws_h) {
    if (threadIdx.x == 0 && blockIdx.x == 0) {
        double S_hat = (double)x[MM];
        double S     = 10.0 * S_hat;
        double alpha = 0.8 * S / (2.0 + S + 1e-12);
        double mmax  = (double)m[MM - 1];
        double vmax  = fabs(alpha) * mmax;
        double cfl   = vmax * 0.002 / (D_DM + 1e-12);
        int N_cfl = (vmax == 0.0 || cfl <= 0.8) ? 1 : (int)ceil(cfl / 0.8);
        double rgv  = 0.8 * S / (2.0 + S + 1e-12) * mmax;
        double gm   = 1.0 / (1.0 - mmax + 1e-12) - 1.0 / (1.0 - 0.45 + 1e-12);
        if (gm < 0.0) gm = 0.0;
        double rate = gm * rgv;
        double rr   = rate * 0.002;
        int N_re = (rate == 0.0 || rr <= 0.5) ? 1 : (int)ceil(rr / 0.5);
        int N = N_cfl > N_re ? N_cfl : N_re;
        if (N < 1) N = 1;
        if (N > NMAX) N = NMAX;
        *ws_N = N;
        *ws_h = (float)(0.002 / (double)N);
    }
}

// ---------------------------------------------------------------------------
// Kernel A (one 1024-thread block): RK4 transport -> n_tmp, then G, v = G*n_tmp,
// uptake reduction and S update. O(M) work, single WGP, LDS for neighbor access.
// ---------------------------------------------------------------------------
__global__ void __launch_bounds__(1024)
substep_local_k(const float* __restrict__ m,
                float* __restrict__ state,      // d_out: [n(8192), S]
                float* __restrict__ ws_v,
                float* __restrict__ ws_nt,
                float* __restrict__ ws_G,
                const int* __restrict__ ws_N,
                const float* __restrict__ ws_h,
                int k) {
    if (k >= *ws_N) return;
    __shared__ float sin_[MM];   // 32 KB stage buffer
    __shared__ float red[32];

    const float h     = *ws_h;
    const float S_hat = state[MM];
    const float S     = F_S_MAX * S_hat;
    const float alpha = F_KS_MAX * S / (F_K_S + S);       // transport alpha (no EPS)
    const int   t     = threadIdx.x;

    float nh[8], mreg[8], accr[8], ks[8];
    #pragma unroll
    for (int q = 0; q < 8; ++q) {
        int i = t + q * 1024;
        nh[q]   = state[i];
        mreg[q] = m[i];
        accr[q] = 0.0f;
        sin_[i] = nh[q];
    }

    const float wgt[4] = {1.0f, 2.0f, 2.0f, 1.0f};
    const float cs[4]  = {0.5f, 0.5f, 1.0f, 0.0f};

    for (int s = 0; s < 4; ++s) {
        __syncthreads();
        #pragma unroll
        for (int q = 0; q < 8; ++q) {
            int i = t + q * 1024;
            float wi   = alpha * mreg[q] * sin_[i];
            float wim1 = (i == 0) ? 0.0f : alpha * m[i - 1] * sin_[i - 1];
            float T;
            if (i == MM - 1) {
                // w[last] zeroed before diff, then flux_out/dm added back
                T = (wim1 + wi) * F_INV_DM;
            } else {
                T = -(wi - wim1) * F_INV_DM;
            }
            ks[q]    = T;
            accr[q] += wgt[s] * T;
        }
        __syncthreads();
        if (s < 3) {
            #pragma unroll
            for (int q = 0; q < 8; ++q) {
                int i = t + q * 1024;
                sin_[i] = nh[q] + cs[s] * h * ks[q];
            }
        }
    }

    const float rg_coef = F_KS_MAX * S / (F_K_S + S + F_EPS);
    const float gm_sub  = 1.0f / (F_M_MAX - F_M_MIN + F_EPS);
    float upt = 0.0f;
    #pragma unroll
    for (int q = 0; q < 8; ++q) {
        int i = t + q * 1024;
        float ntmp = nh[q] + (h / 6.0f) * accr[q];
        float rg   = rg_coef * mreg[q];
        float gm   = fmaxf(1.0f / (F_M_MAX - mreg[q] + F_EPS) - gm_sub, 0.0f);
        float G    = gm * rg;
        ws_nt[i] = ntmp;
        ws_G[i]  = G;
        ws_v[i]  = G * ntmp;
        upt += rg * ntmp;
    }
    // wave32 + cross-wave reduction of uptake
    #pragma unroll
    for (int off = 16; off > 0; off >>= 1) upt += __shfl_xor(upt, off);
    if ((t & 31) == 0) red[t >> 5] = upt;
    __syncthreads();
    if (t < 32) {
        float r = red[t];
        #pragma unroll
        for (int off = 16; off > 0; off >>= 1) r += __shfl_xor(r, off);
        if (t == 0) {
            float uptake = r * F_DM;
            float Snext  = S_hat + h * (-F_KAPPA * uptake);
            state[MM] = fmaxf(Snext, 0.0f);
        }
    }
}

// ---------------------------------------------------------------------------
// Kernel B: gain = 2*dm * (v @ P), then implicit update of n into state.
// One wave owns 16 output columns; K swept in chunks of 4 with
// V_WMMA_F32_16X16X4_F32; A row 0 carries v (branch-free arithmetic mask),
// B carries the 4x16 P tile. Outer loop = 64-row chunk with one prefetch;
// inner body = 16 WMMAs fully unrolled so loads can run ahead of the waits.
// Bandwidth-bound: streams all 256 MB of P once per substep.
// ---------------------------------------------------------------------------
__global__ void __launch_bounds__(256)
substep_gain_k(const float* __restrict__ P,
               const float* __restrict__ ws_v,
               const float* __restrict__ ws_nt,
               const float* __restrict__ ws_G,
               float* __restrict__ state,
               const int* __restrict__ ws_N,
               const float* __restrict__ ws_h,
               int k) {
    if (k >= *ws_N) return;
    __shared__ float vlds[MM];   // 32 KB: whole v vector
    for (int i = threadIdx.x; i < MM; i += 256) vlds[i] = ws_v[i];

    const int wave = threadIdx.x >> 5;          // 0..7
    const int lane = threadIdx.x & 31;
    const int j0   = (blockIdx.x * 8 + wave) * 16;
    const int half = lane >> 4;                 // 0: K=0,1 | 1: K=2,3
    const int lcol = lane & 15;
    // Only A row M=0 is non-zero -> lanes 0 and 16 carry v; others masked to 0.
    const float fmask = (lcol == 0) ? 1.0f : 0.0f;

    const float* Pbase = P + (size_t)(2 * half) * MM + (j0 + lcol);
    const float h = *ws_h;

    __syncthreads();

    v8f acc = {0.f, 0.f, 0.f, 0.f, 0.f, 0.f, 0.f, 0.f};

    for (int c0 = 0; c0 < MM; c0 += 64) {
        if (c0 + 64 < MM)
            __builtin_prefetch(Pbase + (size_t)(c0 + 64) * MM, 0, 0); // stream-ahead
        #pragma unroll
        for (int u = 0; u < 64; u += 4) {
            const int i0 = c0 + u;
            // A (16x4): LDS broadcast read (lanes 0-15 one addr, 16-31 another),
            // arithmetic mask instead of divergent exec.
            v2f a;
            a.x = fmask * vlds[i0 + 2 * half];
            a.y = fmask * vlds[i0 + 2 * half + 1];
            // B (4x16): lanes 0-15 rows i0+0/i0+1, lanes 16-31 rows i0+2/i0+3
            const float* Pp = Pbase + (size_t)i0 * MM;
            v2f b;
            b.x = Pp[0];
            b.y = Pp[MM];
            acc = __builtin_amdgcn_wmma_f32_16x16x4_f32(
                false, a, false, b, (short)0, acc, false, false);
        }
    }

    // D row M=0 lives in acc[0] on lanes 0..15 (N = lane)
    if (lane < 16) {
        int j = j0 + lane;
        float gain = 2.0f * F_DM * acc[0];
        float nt   = ws_nt[j];
        float G    = ws_G[j];
        float nn   = (nt + h * gain) / (1.0f + h * G);
        state[j]   = fmaxf(nn, 0.0f);
    }
}

// ---------------------------------------------------------------------------
extern "C" void kernel_launch(void* const* d_in, const int* in_sizes, int n_in,
                              void* d_out, int out_size, void* d_ws, size_t ws_size,
                              hipStream_t stream) {
    const float* x = (const float*)d_in[0];   // (M+1,)
    const float* m = (const float*)d_in[1];   // (M,)
    const float* P = (const float*)d_in[2];   // (M, M) row-major

    float* state = (float*)d_out;             // state evolves in place in d_out
    float* wsf   = (float*)d_ws;
    float* ws_v  = wsf;
    float* ws_nt = wsf + MM;
    float* ws_G  = wsf + 2 * MM;
    int*   ws_N  = (int*)(wsf + 3 * MM);
    float* ws_h  = wsf + 3 * MM + 1;

    hipMemcpyAsync(state, x, (MM + 1) * sizeof(float),
                   hipMemcpyDeviceToDevice, stream);
    compute_substeps_k<<<1, 1, 0, stream>>>(x, m, ws_N, ws_h);

    for (int k = 0; k < NMAX; ++k) {
        substep_local_k<<<1, 1024, 0, stream>>>(m, state, ws_v, ws_nt, ws_G,
                                                ws_N, ws_h, k);
        substep_gain_k<<<MM / 128, 256, 0, stream>>>(P, ws_v, ws_nt, ws_G,
                                                     state, ws_N, ws_h, k);
    }
}